// VariationalCircuit_91199335563769
// MI455X (gfx1250) — compile-verified
//
#include <hip/hip_runtime.h>
#include <math.h>

typedef __attribute__((ext_vector_type(2))) float v2f;
typedef __attribute__((ext_vector_type(4))) float v4f;
typedef __attribute__((ext_vector_type(8))) float v8f;

#define NQ 4
#define NL 2

// ---------------------------------------------------------------------------
// Prep kernel: build the fixed 16x16 complex matrix V = (layer ops) from
// weights, then store it in WMMA A-fragment order into d_ws.
// Wire w <-> bit w (LSB) consistently; outputs are invariant to this
// relabeling since every op (gates, CNOT ring, per-wire expvals) is defined
// per-wire.
// Launch: 1 block, 32 threads (one wave).
// ---------------------------------------------------------------------------
__global__ void vqc_prep(const float* __restrict__ weights,
                         float* __restrict__ wsA)
{
    __shared__ float vre[16][16];
    __shared__ float vim[16][16];
    const int lane = threadIdx.x;      // 0..31
    const int col  = lane & 15;

    // lane (0..15) simulates basis column `col`; state kept in registers.
    float str[16], sti[16];
#pragma unroll
    for (int i = 0; i < 16; ++i) { str[i] = (i == col) ? 1.f : 0.f; sti[i] = 0.f; }

#pragma unroll
    for (int l = 0; l < NL; ++l) {
        // Rot(phi,theta,omega) = RZ(omega) RY(theta) RZ(phi) on each wire
#pragma unroll
        for (int w = 0; w < NQ; ++w) {
            float phi = weights[(l * NQ + w) * 3 + 0];
            float th  = weights[(l * NQ + w) * 3 + 1];
            float om  = weights[(l * NQ + w) * 3 + 2];
            float c, s, ca, sa, cb, sb;
            __sincosf(0.5f * th, &s, &c);
            __sincosf(0.5f * (phi + om), &sa, &ca);
            __sincosf(0.5f * (phi - om), &sb, &cb);
            // U00 = ep*c = (c*ca, -c*sa)      ep = exp(-i(phi+om)/2)
            // U01 = -conj(em)*s = (-s*cb, -s*sb)
            // U10 = em*s = (s*cb, -s*sb)      em = exp(-i(phi-om)/2)
            // U11 = conj(ep)*c = (c*ca, c*sa)
            const float u00r =  c * ca, u00i = -c * sa;
            const float u01r = -s * cb, u01i = -s * sb;
            const float u10r =  s * cb, u10i = -s * sb;
            const float u11r =  c * ca, u11i =  c * sa;
#pragma unroll
            for (int i = 0; i < 16; ++i) {
                if (((i >> w) & 1) == 0) {
                    const int i1 = i | (1 << w);
                    const float r0 = str[i],  q0 = sti[i];
                    const float r1 = str[i1], q1 = sti[i1];
                    str[i]  = u00r * r0 - u00i * q0 + u01r * r1 - u01i * q1;
                    sti[i]  = u00r * q0 + u00i * r0 + u01r * q1 + u01i * r1;
                    str[i1] = u10r * r0 - u10i * q0 + u11r * r1 - u11i * q1;
                    sti[i1] = u10r * q0 + u10i * r0 + u11r * q1 + u11i * r1;
                }
            }
        }
        // CNOT ring: control w, target (w+1)%4
#pragma unroll
        for (int w = 0; w < NQ; ++w) {
            const int t = (w + 1) & 3;
#pragma unroll
            for (int i = 0; i < 16; ++i) {
                if ((((i >> w) & 1) == 1) && (((i >> t) & 1) == 0)) {
                    const int i1 = i ^ (1 << t);
                    float tr = str[i], ti = sti[i];
                    str[i]  = str[i1]; sti[i]  = sti[i1];
                    str[i1] = tr;      sti[i1] = ti;
                }
            }
        }
    }

    if (lane < 16) {
#pragma unroll
        for (int k = 0; k < 16; ++k) { vre[k][col] = str[k]; vim[k][col] = sti[k]; }
    }
    __syncthreads();

    // A-fragment order for V_WMMA_F32_16X16X4_F32:
    //   lane L: m = L%16, hi = L/16; chunk kc, VGPR j -> element V[m][4kc+j+2hi]
    const int m = lane & 15, hi = lane >> 4;
    float frag[16];
#pragma unroll
    for (int kc = 0; kc < 4; ++kc) {
#pragma unroll
        for (int j = 0; j < 2; ++j) {
            const int k = 4 * kc + j + 2 * hi;
            frag[2 * kc + j]     = vre[m][k];
            frag[8 + 2 * kc + j] = vim[m][k];
        }
    }
#pragma unroll
    for (int i = 0; i < 16; ++i) wsA[lane * 16 + i] = frag[i];
}

// ---------------------------------------------------------------------------
// Main kernel: one 16-sample tile per wave-iteration.
//   S[k][n] = prod_w (bit_w(k) ? sin(h_w) : cos(h_w)),  h_w = tanh(x)*pi/2
//   Y_re = Re(V) @ S,  Y_im = Im(V) @ S   via 8x v_wmma_f32_16x16x4_f32
//   out_w = sum_k sign_w(k) (Y_re^2 + Y_im^2)
// ---------------------------------------------------------------------------
__global__ void vqc_main(const float* __restrict__ x,
                         const float* __restrict__ wsA,
                         float* __restrict__ out, int batch)
{
    const float PI_H = 1.57079632679489661923f;  // pi/2
    const int lane   = threadIdx.x & 31;
    const int wave   = blockIdx.x * (blockDim.x >> 5) + (threadIdx.x >> 5);
    const int nwaves = gridDim.x * (blockDim.x >> 5);
    const int ntiles = (batch + 15) >> 4;
    const int hi     = lane >> 4;

    // Load the per-lane A fragments once (re: kc0..3, im: kc0..3).
    v2f aRe[4], aIm[4];
    {
        const v4f* p = (const v4f*)(wsA + lane * 16);
        v4f f0 = p[0], f1 = p[1], f2 = p[2], f3 = p[3];
        aRe[0] = v2f{f0.x, f0.y}; aRe[1] = v2f{f0.z, f0.w};
        aRe[2] = v2f{f1.x, f1.y}; aRe[3] = v2f{f1.z, f1.w};
        aIm[0] = v2f{f2.x, f2.y}; aIm[1] = v2f{f2.z, f2.w};
        aIm[2] = v2f{f3.x, f3.y}; aIm[3] = v2f{f3.z, f3.w};
    }

    for (int t = wave; t < ntiles; t += nwaves) {
        const int sample = t * 16 + (lane & 15);
        const int ls     = (sample < batch) ? sample : 0;

        v4f xv = *(const v4f*)(x + (size_t)ls * 8);   // first 4 of 8 features
        const float h0 = tanhf(xv.x) * PI_H;
        const float h1 = tanhf(xv.y) * PI_H;
        const float h2 = tanhf(xv.z) * PI_H;
        const float h3 = tanhf(xv.w) * PI_H;
        float c0, s0, c1, s1, c2, s2, c3, s3;
        __sincosf(h0, &s0, &c0); __sincosf(h1, &s1, &c1);
        __sincosf(h2, &s2, &c2); __sincosf(h3, &s3, &c3);

        // t01[b0 + 2*b1], t23[b2 + 2*b3]
        const float t01_0 = c0 * c1, t01_1 = s0 * c1;
        const float t01_2 = c0 * s1, t01_3 = s0 * s1;
        const float t23[4] = {c2 * c3, s2 * c3, c2 * s3, s2 * s3};
        // B fragment element (chunk kc, VGPR j): S[4kc + j + 2hi][n]
        //   = t01[j + 2*hi] * t23[kc]
        const float u0 = hi ? t01_2 : t01_0;
        const float u1 = hi ? t01_3 : t01_1;

        v8f accRe = {}; v8f accIm = {};
#pragma unroll
        for (int kc = 0; kc < 4; ++kc) {
            v2f b = v2f{u0 * t23[kc], u1 * t23[kc]};
            accRe = __builtin_amdgcn_wmma_f32_16x16x4_f32(
                false, aRe[kc], false, b, (short)0, accRe, false, false);
            accIm = __builtin_amdgcn_wmma_f32_16x16x4_f32(
                false, aIm[kc], false, b, (short)0, accIm, false, false);
        }

        // Probabilities: lane holds rows kappa = r + 8*hi, sample n = lane%16.
        float p[8];
#pragma unroll
        for (int r = 0; r < 8; ++r)
            p[r] = accRe[r] * accRe[r] + accIm[r] * accIm[r];

        float o0 = (p[0] - p[1]) + (p[2] - p[3]) + (p[4] - p[5]) + (p[6] - p[7]);
        float o1 = (p[0] + p[1]) - (p[2] + p[3]) + (p[4] + p[5]) - (p[6] + p[7]);
        float o2 = (p[0] + p[1]) + (p[2] + p[3]) - (p[4] + p[5]) - (p[6] + p[7]);
        float tt = (p[0] + p[1]) + (p[2] + p[3]) + (p[4] + p[5]) + (p[6] + p[7]);
        float o3 = hi ? -tt : tt;

        // Combine the two lane halves (kappa bit3): SWAPX16 ds_swizzle.
        o0 += __int_as_float(__builtin_amdgcn_ds_swizzle(__float_as_int(o0), 0x401F));
        o1 += __int_as_float(__builtin_amdgcn_ds_swizzle(__float_as_int(o1), 0x401F));
        o2 += __int_as_float(__builtin_amdgcn_ds_swizzle(__float_as_int(o2), 0x401F));
        o3 += __int_as_float(__builtin_amdgcn_ds_swizzle(__float_as_int(o3), 0x401F));

        if (lane < 16 && sample < batch) {
            v4f o = v4f{o0, o1, o2, o3};
            *(v4f*)(out + (size_t)sample * 4) = o;
        }
    }
}

extern "C" void kernel_launch(void* const* d_in, const int* in_sizes, int n_in,
                              void* d_out, int out_size, void* d_ws, size_t ws_size,
                              hipStream_t stream)
{
    const float* x       = (const float*)d_in[0];
    const float* weights = (const float*)d_in[1];
    float* out = (float*)d_out;
    float* wsA = (float*)d_ws;          // uses 512 floats (2 KB)

    const int batch  = in_sizes[0] / 8; // x is [B, 8]
    const int ntiles = (batch + 15) >> 4;

    vqc_prep<<<1, 32, 0, stream>>>(weights, wsA);

    const int threads = 256;            // 8 waves per block
    const int wpb = threads / 32;
    int blocks = (ntiles + wpb - 1) / wpb;
    if (blocks > 2048) blocks = 2048;   // grid-stride over tiles
    if (blocks < 1) blocks = 1;
    vqc_main<<<blocks, threads, 0, stream>>>(x, wsA, out, batch);
}